// LightGCN_42597485642223
// MI455X (gfx1250) — compile-verified
//
#include <hip/hip_runtime.h>

// ---------------------------------------------------------------------------
// LightGCN propagation on MI455X (gfx1250).
// Memory/atomic-bound SpMM: 75000x128 f32 embeddings (38.4MB -> resident in
// 192MB L2), 1.6M edges, 3 layers. Wave32: one wave per edge, lane = float4
// chunk of the 128-wide row. Per-edge norm weights precomputed once
// (layer-invariant); edge indices + weights staged to LDS with the gfx1250
// async-to-LDS engine (ASYNCcnt path).
// ---------------------------------------------------------------------------

constexpr int   D_    = 128;
constexpr float SCALE = 0.25f;   // 1/(NUM_LAYERS+1)
constexpr int   LAYERS = 3;
constexpr int   EPB   = 256;     // edges per block in SpMM (== blockDim)

#define AS_GLOBAL __attribute__((address_space(1)))
#define AS_SHARED __attribute__((address_space(3)))

#if defined(__HIP_DEVICE_COMPILE__) && \
    __has_builtin(__builtin_amdgcn_global_load_async_to_lds_b32) && \
    __has_builtin(__builtin_amdgcn_s_wait_asynccnt)
#define USE_ASYNC_LDS 1
#else
#define USE_ASYNC_LDS 0
#endif

__global__ __launch_bounds__(256) void k_zero_f(float* __restrict__ p, int n) {
    int i = blockIdx.x * blockDim.x + threadIdx.x;
    if (i < n) p[i] = 0.0f;
}

__global__ __launch_bounds__(256) void k_zero_f4(float4* __restrict__ p, int n4) {
    int i = blockIdx.x * blockDim.x + threadIdx.x;
    if (i < n4) p[i] = float4{0.f, 0.f, 0.f, 0.f};
}

__global__ __launch_bounds__(256) void k_degree(const int* __restrict__ src,
                                                float* __restrict__ deg, int E) {
    int e = blockIdx.x * blockDim.x + threadIdx.x;
    if (e < E) {
        __hip_atomic_fetch_add(&deg[src[e]], 1.0f,
                               __ATOMIC_RELAXED, __HIP_MEMORY_SCOPE_AGENT);
    }
}

// deg -> dinv in place: deg>0 ? rsqrt(deg) : 0
__global__ __launch_bounds__(256) void k_dinv(float* __restrict__ deg, int n) {
    int i = blockIdx.x * blockDim.x + threadIdx.x;
    if (i < n) {
        float d = deg[i];
        deg[i] = (d > 0.0f) ? __frsqrt_rn(d) : 0.0f;
    }
}

// Per-edge weight (layer-invariant): w[e] = dinv[src[e]] * dinv[dst[e]]
__global__ __launch_bounds__(256) void k_edgew(const int* __restrict__ src,
                                               const int* __restrict__ dst,
                                               const float* __restrict__ dinv,
                                               float* __restrict__ w, int E) {
    int e = blockIdx.x * blockDim.x + threadIdx.x;
    if (e < E) w[e] = dinv[src[e]] * dinv[dst[e]];
}

// x = concat(user_emb, item_emb); avg = SCALE * x   (float4 granularity)
__global__ __launch_bounds__(256) void k_init(const float4* __restrict__ ue,
                                              const float4* __restrict__ ie,
                                              float4* __restrict__ x,
                                              float4* __restrict__ avg,
                                              int nu4, int n4) {
    int i = blockIdx.x * blockDim.x + threadIdx.x;
    if (i < n4) {
        float4 v = (i < nu4) ? ue[i] : ie[i - nu4];
        x[i] = v;
        avg[i] = float4{v.x * SCALE, v.y * SCALE, v.z * SCALE, v.w * SCALE};
    }
}

__global__ __launch_bounds__(256) void k_axpy(float4* __restrict__ avg,
                                              const float4* __restrict__ x, int n4) {
    int i = blockIdx.x * blockDim.x + threadIdx.x;
    if (i < n4) {
        float4 a = avg[i];
        float4 v = x[i];
        avg[i] = float4{a.x + v.x * SCALE, a.y + v.y * SCALE,
                        a.z + v.z * SCALE, a.w + v.w * SCALE};
    }
}

// One wave (32 lanes) per edge; lane l owns floats [4l, 4l+4) of the row.
// Edge indices + weights for the block are staged into LDS via async-to-LDS;
// each wave consumes the 32 edges it staged (no cross-wave barrier needed).
__global__ __launch_bounds__(256) void k_spmm(const int* __restrict__ src,
                                              const int* __restrict__ dst,
                                              const float* __restrict__ ew,
                                              const float* __restrict__ xin,
                                              float* __restrict__ xout, int E) {
    __shared__ int   s_src[EPB];
    __shared__ int   s_dst[EPB];
    __shared__ float s_w[EPB];

    const int tid  = threadIdx.x;         // == LDS slot
    const int wave = tid >> 5;
    const int lane = tid & 31;
    const int base = blockIdx.x * EPB;
    const int eg   = base + tid;
    const int egc  = (eg < E) ? eg : (E - 1);   // keep staging address valid

#if USE_ASYNC_LDS
    __builtin_amdgcn_global_load_async_to_lds_b32(
        (AS_GLOBAL int*)(src + egc),
        (AS_SHARED int*)(&s_src[tid]), 0, 0);
    __builtin_amdgcn_global_load_async_to_lds_b32(
        (AS_GLOBAL int*)(dst + egc),
        (AS_SHARED int*)(&s_dst[tid]), 0, 0);
    __builtin_amdgcn_global_load_async_to_lds_b32(
        (AS_GLOBAL int*)(ew + egc),
        (AS_SHARED int*)(&s_w[tid]), 0, 0);
    __builtin_amdgcn_s_wait_asynccnt(0);
#else
    s_src[tid] = src[egc];
    s_dst[tid] = dst[egc];
    s_w[tid]   = ew[egc];
#endif

    // Prefetch the gather row of the edge this thread staged (512B row).
    {
        const int d = s_dst[tid];
        __builtin_prefetch(xin + (size_t)d * D_, 0, 3);
        __builtin_prefetch(xin + (size_t)d * D_ + 64, 0, 3);
    }

    const int remain = E - (base + (wave << 5));
    const int kmax   = remain < 32 ? (remain < 0 ? 0 : remain) : 32;

    for (int k = 0; k < kmax; ++k) {
        const int   s = s_src[(wave << 5) + k];   // LDS broadcasts
        const int   d = s_dst[(wave << 5) + k];
        const float w = s_w[(wave << 5) + k];

        const float4 v = ((const float4*)(xin + (size_t)d * D_))[lane];
        float* o = xout + (size_t)s * D_ + (lane << 2);

        __hip_atomic_fetch_add(o + 0, v.x * w, __ATOMIC_RELAXED, __HIP_MEMORY_SCOPE_AGENT);
        __hip_atomic_fetch_add(o + 1, v.y * w, __ATOMIC_RELAXED, __HIP_MEMORY_SCOPE_AGENT);
        __hip_atomic_fetch_add(o + 2, v.z * w, __ATOMIC_RELAXED, __HIP_MEMORY_SCOPE_AGENT);
        __hip_atomic_fetch_add(o + 3, v.w * w, __ATOMIC_RELAXED, __HIP_MEMORY_SCOPE_AGENT);
    }
}

extern "C" void kernel_launch(void* const* d_in, const int* in_sizes, int n_in,
                              void* d_out, int out_size, void* d_ws, size_t ws_size,
                              hipStream_t stream) {
    const float* ue   = (const float*)d_in[0];
    const float* ie   = (const float*)d_in[1];
    const int*   esrc = (const int*)d_in[2];
    const int*   edst = (const int*)d_in[3];

    const int Usz = in_sizes[0];        // U*D floats
    const int Isz = in_sizes[1];        // I*D floats
    const int E   = in_sizes[2];        // 1,600,000
    const int Nd  = Usz + Isz;          // N*D == out_size
    const int N   = Nd / D_;
    const int n4  = Nd / 4;
    const int nu4 = Usz / 4;

    // Workspace layout: dinv[N] | w[E] | xA[N*D] | xB[N*D]
    float* deg = (float*)d_ws;
    float* ew  = deg + ((N + 255) & ~255);
    float* xA  = ew + ((E + 255) & ~255);
    float* xB  = xA + Nd;
    float* avg = (float*)d_out;

    const int T = 256;
    auto cdiv = [](int a, int b) { return (a + b - 1) / b; };

    // dinv + per-edge weights (layer-invariant)
    k_zero_f<<<cdiv(N, T), T, 0, stream>>>(deg, N);
    k_degree<<<cdiv(E, T), T, 0, stream>>>(esrc, deg, E);
    k_dinv<<<cdiv(N, T), T, 0, stream>>>(deg, N);
    k_edgew<<<cdiv(E, T), T, 0, stream>>>(esrc, edst, deg, ew, E);

    // x0 and avg = x0/4
    k_init<<<cdiv(n4, T), T, 0, stream>>>((const float4*)ue, (const float4*)ie,
                                          (float4*)xA, (float4*)avg, nu4, n4);

    for (int l = 0; l < LAYERS; ++l) {
        k_zero_f4<<<cdiv(n4, T), T, 0, stream>>>((float4*)xB, n4);
        k_spmm<<<cdiv(E, EPB), EPB, 0, stream>>>(esrc, edst, ew, xA, xB, E);
        k_axpy<<<cdiv(n4, T), T, 0, stream>>>((float4*)avg, (const float4*)xB, n4);
        float* t = xA; xA = xB; xB = t;
    }
}